// StackedDecoder_63050119906015
// MI455X (gfx1250) — compile-verified
//
#include <hip/hip_runtime.h>
#include <hip/hip_bf16.h>
#include <stdint.h>

// ---------------------------------------------------------------------------
// StackedDecoder (2-layer graph GRU) for MI455X / gfx1250
//  - fp32 WMMA 16x16x4 for all dense projections (matches reference precision;
//    the pipeline is aggregation/L2-bound, so fp32 matrix ops cost nothing
//    end-to-end and keep exact semantics)
//  - weight tiles staged into LDS with global_load_async_to_lds_b128
//    (ASYNCcnt) and consumed via ds_load
//  - L2-resident fp32 atomic scatter-adds for the graph aggregations
// ---------------------------------------------------------------------------

typedef float v2f __attribute__((ext_vector_type(2)));
typedef float v8f __attribute__((ext_vector_type(8)));

enum { C_H = 256, C_FIN = 128, C_FOUT = 128, KMAX = 256 };

// ---------------------------------------------------------------------------
// Zero fill (grid-stride)
// ---------------------------------------------------------------------------
__global__ void k_zero(float* __restrict__ p, long n) {
    long i = (long)blockIdx.x * blockDim.x + threadIdx.x;
    long stride = (long)gridDim.x * blockDim.x;
    for (; i < n; i += stride) p[i] = 0.0f;
}

// ---------------------------------------------------------------------------
// Edge-parallel segment-sum: out[dst[e], :] += feat[src[e], :]
// float4 per thread; consecutive threads cover consecutive features of an edge
// -> coalesced gathers, L2-resident fp32 atomics for the scatter.
// ---------------------------------------------------------------------------
__global__ void k_agg(const float* __restrict__ feat,
                      const int* __restrict__ src,
                      const int* __restrict__ dst,
                      float* __restrict__ out,
                      int Fv, int F, long total) {
    long i = (long)blockIdx.x * blockDim.x + threadIdx.x;
    if (i >= total) return;
    long e = i / Fv;
    int  v = (int)(i - e * Fv) * 4;
    const float4 val = *(const float4*)&feat[(size_t)src[e] * F + v];
    float* o = &out[(size_t)dst[e] * F + v];
    unsafeAtomicAdd(o + 0, val.x);
    unsafeAtomicAdd(o + 1, val.y);
    unsafeAtomicAdd(o + 2, val.z);
    unsafeAtomicAdd(o + 3, val.w);
}

// ---------------------------------------------------------------------------
// pre[g][n][col] = bx[g][col] + bh[g][col]    (3 gates)
// ---------------------------------------------------------------------------
__global__ void k_init_pre(float* __restrict__ pre,
                           const float* __restrict__ bx,
                           const float* __restrict__ bh,
                           long NH, long total) {
    long i = (long)blockIdx.x * blockDim.x + threadIdx.x;
    if (i >= total) return;
    long g   = i / NH;
    int  col = (int)(i % (long)C_H);
    pre[i] = bx[g * C_H + col] + bh[g * C_H + col];
}

// out[n][col] = bias[col]
__global__ void k_init_bias(float* __restrict__ out,
                            const float* __restrict__ bias,
                            int F, long total) {
    long i = (long)blockIdx.x * blockDim.x + threadIdx.x;
    if (i >= total) return;
    out[i] = bias[(int)(i % F)];
}

// ---------------------------------------------------------------------------
// C[M,Nc] += A[M,K] @ B[K,Nc]   via V_WMMA_F32_16X16X4_F32
//
// Block = 256 threads = 8 waves. Each block owns a K x 32 column slice of B,
// staged once into LDS with global_load_async_to_lds_b128 (ASYNCcnt), then
// shared by all 8 waves (8 M-tiles x 2 N-tiles each). Inner loop per wave:
// one global_load_b64 of A feeds two WMMAs whose B operands come from LDS.
// M, K, Nc are multiples of 16/32 here (M=20000, K in {128,256}, Nc in {128,256}).
// ---------------------------------------------------------------------------
__global__ __launch_bounds__(256)
void k_wmma_gemm_acc(const float* __restrict__ A,
                     const float* __restrict__ B,
                     float* __restrict__ C,
                     int M, int K, int Nc) {
    __shared__ float sB[KMAX * 32];            // <= 32 KB: B[k][col0+0..31]

    const int nBlocks = Nc >> 5;               // 32-column slices
    const int nb   = blockIdx.x % nBlocks;
    const int mb   = blockIdx.x / nBlocks;
    const int col0 = nb << 5;

    // ---- async-stage B[K x 32] into LDS (row-major, row stride 32 floats) ----
    {
        const int chunks = K << 3;             // 16-byte chunks: K rows * 8
        for (int c = threadIdx.x; c < chunks; c += 256) {
            int k = c >> 3;
            int f = (c & 7) << 2;              // float offset within row
            const float* gsrc = B + (size_t)k * Nc + col0 + f;
            uint32_t lds = (uint32_t)(uintptr_t)&sB[(k << 5) + f];
            uint64_t ga  = (uint64_t)(uintptr_t)gsrc;
            asm volatile("global_load_async_to_lds_b128 %0, %1, off"
                         :: "v"(lds), "v"(ga) : "memory");
        }
        asm volatile("s_wait_asynccnt 0x0" ::: "memory");
    }
    __syncthreads();

    // ---- compute: 8 waves x (16 rows x 32 cols) each ----
    const int wid = threadIdx.x >> 5;
    const int tm  = (mb << 3) + wid;
    if (tm < (M >> 4)) {                       // wave-uniform guard: EXEC all-ones
        const int lane = threadIdx.x & 31;
        const int half = lane >> 4;            // 0: K=0,1 / rows 0..7 ; 1: K=2,3 / rows 8..15
        const int l16  = lane & 15;

        const int m0 = (tm << 4) + (half << 3);
        const int n0 = col0 + l16;

        v8f acc0, acc1;
#pragma unroll
        for (int r = 0; r < 8; ++r) {
            acc0[r] = C[(size_t)(m0 + r) * Nc + n0];
            acc1[r] = C[(size_t)(m0 + r) * Nc + n0 + 16];
        }

        const float* Arow = A + (size_t)((tm << 4) + l16) * K + (half << 1);

#pragma unroll 4
        for (int k0 = 0; k0 < K; k0 += 4) {
            v2f a = *(const v2f*)(Arow + k0);  // 8B aligned: offsets even
            const int ka = k0 + (half << 1);
            v2f b0, b1;
            b0.x = sB[(ka << 5) + l16];
            b0.y = sB[((ka + 1) << 5) + l16];
            b1.x = sB[(ka << 5) + 16 + l16];
            b1.y = sB[((ka + 1) << 5) + 16 + l16];
            acc0 = __builtin_amdgcn_wmma_f32_16x16x4_f32(
                false, a, false, b0, (short)0, acc0, false, false);
            acc1 = __builtin_amdgcn_wmma_f32_16x16x4_f32(
                false, a, false, b1, (short)0, acc1, false, false);
        }

#pragma unroll
        for (int r = 0; r < 8; ++r) {
            C[(size_t)(m0 + r) * Nc + n0]      = acc0[r];
            C[(size_t)(m0 + r) * Nc + n0 + 16] = acc1[r];
        }
    }
}

// ---------------------------------------------------------------------------
// r = sigmoid(pre0) (in place), u = sigmoid(pre1) (in place), rh = r * h
// ---------------------------------------------------------------------------
__global__ void k_gates(float* __restrict__ pre0,
                        float* __restrict__ pre1,
                        const float* __restrict__ h,
                        float* __restrict__ rh, long NH) {
    long i = (long)blockIdx.x * blockDim.x + threadIdx.x;
    if (i >= NH) return;
    float r = 1.0f / (1.0f + __expf(-pre0[i]));
    float u = 1.0f / (1.0f + __expf(-pre1[i]));
    pre0[i] = r;
    pre1[i] = u;
    rh[i]   = r * h[i];
}

// c = tanh(pre2);  h = u*h + (1-u)*c
__global__ void k_update(const float* __restrict__ pre2,
                         const float* __restrict__ u,
                         float* __restrict__ h, long NH) {
    long i = (long)blockIdx.x * blockDim.x + threadIdx.x;
    if (i >= NH) return;
    float c  = tanhf(pre2[i]);
    float uu = u[i];
    h[i] = uu * h[i] + (1.0f - uu) * c;
}

// ---------------------------------------------------------------------------
// Host-side helpers
// ---------------------------------------------------------------------------
static inline int cdivl(long a, int b) { return (int)((a + b - 1) / b); }

static void launch_zero(float* p, long n, hipStream_t s) {
    int blocks = cdivl(n, 256);
    if (blocks > 1048576) blocks = 1048576;   // grid-stride covers the rest
    k_zero<<<blocks, 256, 0, s>>>(p, n);
}

static void launch_agg(const float* feat, const int* src, const int* dst,
                       float* out, int F, int E, hipStream_t s) {
    int  Fv    = F / 4;
    long total = (long)E * Fv;
    k_agg<<<cdivl(total, 256), 256, 0, s>>>(feat, src, dst, out, Fv, F, total);
}

static void launch_gemm_acc(const float* A, const float* B, float* C,
                            int M, int K, int Nc, hipStream_t s) {
    int mBlocks = (M / 16 + 7) / 8;           // 8 M-tiles (128 rows) per block
    int nBlocks = Nc / 32;                    // 32 columns per block
    k_wmma_gemm_acc<<<mBlocks * nBlocks, 256, 0, s>>>(A, B, C, M, K, Nc);
}

// One GRU cell: h updated in place.
// xin: [N,Kx] aggregated input; Wx: [3,Kx,H]; Wh: [3,H,H]
static void gru_layer(const float* xin, int Kx,
                      const float* Wx, const float* bx,
                      const float* Wh, const float* bh,
                      float* h,
                      float* aggB, float* aggC, float* pre,
                      const int* src, const int* dst, int E, int N,
                      hipStream_t s) {
    const long NH = (long)N * C_H;

    // h_agg = segment_sum(h)
    launch_zero(aggB, NH, s);
    launch_agg(h, src, dst, aggB, C_H, E, s);

    // pre[g] = bx[g] + bh[g]
    k_init_pre<<<cdivl(3 * NH, 256), 256, 0, s>>>(pre, bx, bh, NH, 3 * NH);

    // pre[g] += xin @ Wx[g]
    for (int g = 0; g < 3; ++g)
        launch_gemm_acc(xin, Wx + (size_t)g * Kx * C_H, pre + (size_t)g * NH,
                        N, Kx, C_H, s);
    // pre[0,1] += h_agg @ Wh[g]
    for (int g = 0; g < 2; ++g)
        launch_gemm_acc(aggB, Wh + (size_t)g * C_H * C_H, pre + (size_t)g * NH,
                        N, C_H, C_H, s);

    // r, u, rh  (rh overwrites aggB; h_agg no longer needed)
    k_gates<<<cdivl(NH, 256), 256, 0, s>>>(pre, pre + NH, h, aggB, NH);

    // rh_agg = segment_sum(r*h)
    launch_zero(aggC, NH, s);
    launch_agg(aggB, src, dst, aggC, C_H, E, s);

    // pre[2] += rh_agg @ Wh[2]
    launch_gemm_acc(aggC, Wh + 2 * (size_t)C_H * C_H, pre + 2 * NH, N, C_H, C_H, s);

    // h = u*h + (1-u)*tanh(pre2)
    k_update<<<cdivl(NH, 256), 256, 0, s>>>(pre + 2 * NH, pre + NH, h, NH);
}

// ---------------------------------------------------------------------------
extern "C" void kernel_launch(void* const* d_in, const int* in_sizes, int n_in,
                              void* d_out, int out_size, void* d_ws, size_t ws_size,
                              hipStream_t stream) {
    (void)n_in; (void)out_size; (void)ws_size;

    const float* x   = (const float*)d_in[0];   // [T,N,F_IN]
    const float* h0i = (const float*)d_in[1];   // [2,N,H]
    const int*   src = (const int*)d_in[2];     // [E]
    const int*   dst = (const int*)d_in[3];     // [E]
    const float* Wx0 = (const float*)d_in[4];   // [3,F_IN,H]
    const float* bx0 = (const float*)d_in[5];   // [3,H]
    const float* Wh0 = (const float*)d_in[6];   // [3,H,H]
    const float* bh0 = (const float*)d_in[7];   // [3,H]
    const float* Wx1 = (const float*)d_in[8];   // [3,H,H]
    const float* bx1 = (const float*)d_in[9];   // [3,H]
    const float* Wh1 = (const float*)d_in[10];  // [3,H,H]
    const float* bh1 = (const float*)d_in[11];  // [3,H]
    const float* Wo  = (const float*)d_in[12];  // [H,F_OUT]
    const float* bo  = (const float*)d_in[13];  // [F_OUT]

    const int E = in_sizes[2];
    const int N = in_sizes[1] / (2 * C_H);
    const int T = in_sizes[0] / (N * C_FIN);
    const long NH = (long)N * C_H;

    float* outp = (float*)d_out;                // [T,N,F_OUT] ++ [2,N,H]

    // Workspace layout
    float* ws    = (float*)d_ws;
    float* xagg  = ws;                                  // T*N*F_IN
    float* hbuf0 = xagg + (size_t)T * N * C_FIN;        // N*H
    float* hbuf1 = hbuf0 + NH;                          // N*H
    float* aggA  = hbuf1 + NH;                          // N*H (layer-1 input agg)
    float* aggB  = aggA + NH;                           // N*H (h_agg / rh)
    float* aggC  = aggB + NH;                           // N*H (rh_agg)
    float* pre   = aggC + NH;                           // 3*N*H

    // h <- h0
    hipMemcpyAsync(hbuf0, h0i,      NH * sizeof(float), hipMemcpyDeviceToDevice, stream);
    hipMemcpyAsync(hbuf1, h0i + NH, NH * sizeof(float), hipMemcpyDeviceToDevice, stream);

    // x_aggs[t] = segment_sum(x[t]) for all time steps
    launch_zero(xagg, (long)T * N * C_FIN, stream);
    for (int t = 0; t < T; ++t)
        launch_agg(x + (size_t)t * N * C_FIN, src, dst,
                   xagg + (size_t)t * N * C_FIN, C_FIN, E, stream);

    for (int t = 0; t < T; ++t) {
        const float* xa = xagg + (size_t)t * N * C_FIN;

        // Layer 0 (x path K=F_IN)
        gru_layer(xa, C_FIN, Wx0, bx0, Wh0, bh0, hbuf0,
                  aggB, aggC, pre, src, dst, E, N, stream);

        // Layer-1 input = segment_sum(h0_new)
        launch_zero(aggA, NH, stream);
        launch_agg(hbuf0, src, dst, aggA, C_H, E, stream);

        // Layer 1 (x path K=H)
        gru_layer(aggA, C_H, Wx1, bx1, Wh1, bh1, hbuf1,
                  aggB, aggC, pre, src, dst, E, N, stream);

        // out[t] = h1 @ Wo + bo
        float* ot = outp + (size_t)t * N * C_FOUT;
        k_init_bias<<<cdivl((long)N * C_FOUT, 256), 256, 0, stream>>>(
            ot, bo, C_FOUT, (long)N * C_FOUT);
        launch_gemm_acc(hbuf1, Wo, ot, N, C_H, C_FOUT, stream);
    }

    // Final hidden states -> tail of d_out
    float* htail = outp + (size_t)T * N * C_FOUT;
    hipMemcpyAsync(htail,      hbuf0, NH * sizeof(float), hipMemcpyDeviceToDevice, stream);
    hipMemcpyAsync(htail + NH, hbuf1, NH * sizeof(float), hipMemcpyDeviceToDevice, stream);
}